// GATRelationNet_21706764714080
// MI455X (gfx1250) — compile-verified
//
#include <hip/hip_runtime.h>

#define NEG_SLOPE 0.2f

typedef __attribute__((ext_vector_type(2))) float v2f;
typedef __attribute__((ext_vector_type(8))) float v8f;

static __device__ __forceinline__ v8f wmma4(v2f a, v2f b, v8f c) {
  return __builtin_amdgcn_wmma_f32_16x16x4_f32(
      /*neg_a=*/false, a, /*neg_b=*/false, b,
      /*c_mod=*/(short)0, c, /*reuse_a=*/false, /*reuse_b=*/false);
}

// ---------------------------------------------------------------------------
// Row-major GEMM: D(MxN) = A(MxK) * B(KxN), fp32 exact, V_WMMA_F32_16X16X4_F32.
// Register-blocked: one wave32 owns a 32x32 tile = 2 A-frags x 2 B-frags ->
// 4 accumulators, so each fragment load feeds two WMMAs (6 VMEM per 4 WMMA).
// Per-lane fragment layout (ISA 7.12.2), m = lane&15, h = lane>>4:
//   A frag elem j: A[row][k + 2h + j]
//   B frag elem j: B[k + 2h + j][col]
//   acc elem r   : D[m0 + r + 8h][col]
// Requires N % 32 == 0, K % 4 == 0 (true for all call sites: N=512, K in
// {512,1000}). M ragged: A-row reads clamped in-bounds (stale rows only feed
// discarded D rows), stores row-guarded. Control flow reaching WMMA is
// wave-uniform -> EXEC all-ones.
// ---------------------------------------------------------------------------
__global__ void __launch_bounds__(256) wmma_gemm_f32(
    const float* __restrict__ A, const float* __restrict__ B,
    float* __restrict__ D, int M, int N, int K) {
  const int gwave = (int)((blockIdx.x * blockDim.x + threadIdx.x) >> 5);
  const int lane  = (int)(threadIdx.x & 31);
  const int tN = N >> 5;                  // 32-wide macro columns
  const int tM = (M + 31) >> 5;           // 32-tall macro rows
  const int tm = gwave / tN;
  const int tn = gwave - tm * tN;
  if (tm >= tM) return;                   // wave-uniform guard

  const int m  = lane & 15;
  const int h  = lane >> 4;
  const int m0 = tm << 5;
  const int n0 = tn << 5;

  int r0 = m0 + m;      if (r0 >= M) r0 = M - 1;   // clamp: safe, discarded
  int r1 = m0 + 16 + m; if (r1 >= M) r1 = M - 1;
  const float* __restrict__ A0 = A + (size_t)r0 * K + 2 * h;
  const float* __restrict__ A1 = A + (size_t)r1 * K + 2 * h;
  const float* __restrict__ B0 = B + (size_t)(2 * h) * N + n0 + m;
  const float* __restrict__ B1 = B0 + 16;

  v8f acc00 = {}, acc01 = {}, acc10 = {}, acc11 = {};
#pragma unroll 2
  for (int k = 0; k < K; k += 4) {
    // prefetch B rows ~8 k-steps ahead (global_prefetch_b8)
    if (k + 32 < K) {
      __builtin_prefetch(B0 + (size_t)(k + 32) * N, 0, 1);
      __builtin_prefetch(B1 + (size_t)(k + 32) * N, 0, 1);
    }
    v2f a0, a1, b0, b1;
    a0.x = A0[k];              a0.y = A0[k + 1];
    a1.x = A1[k];              a1.y = A1[k + 1];
    b0.x = B0[(size_t)k * N];  b0.y = B0[(size_t)(k + 1) * N];
    b1.x = B1[(size_t)k * N];  b1.y = B1[(size_t)(k + 1) * N];
    acc00 = wmma4(a0, b0, acc00);
    acc01 = wmma4(a0, b1, acc01);
    acc10 = wmma4(a1, b0, acc10);
    acc11 = wmma4(a1, b1, acc11);
  }

#pragma unroll
  for (int r = 0; r < 8; ++r) {
    const int row = m0 + r + 8 * h;
    if (row < M) {
      D[(size_t)row * N + n0 + m]      = acc00[r];
      D[(size_t)row * N + n0 + 16 + m] = acc01[r];
    }
    const int row2 = row + 16;
    if (row2 < M) {
      D[(size_t)row2 * N + n0 + m]      = acc10[r];
      D[(size_t)row2 * N + n0 + 16 + m] = acc11[r];
    }
  }
}

// ---------------------------------------------------------------------------
// f1[i] = att_h[i,:] . a1 ; f2[i] = att_h[i,:] . a2   (one wave per row)
// ---------------------------------------------------------------------------
__global__ void __launch_bounds__(256) f1f2_kernel(
    const float* __restrict__ att_h, const float* __restrict__ att_a,
    float* __restrict__ f1, float* __restrict__ f2, int Nrows, int H) {
  const int row  = (int)((blockIdx.x * blockDim.x + threadIdx.x) >> 5);
  const int lane = (int)(threadIdx.x & 31);
  if (row >= Nrows) return;
  const float* __restrict__ r = att_h + (size_t)row * H;
  float s1 = 0.f, s2 = 0.f;
  for (int hh = lane; hh < H; hh += 32) {
    float v = r[hh];
    s1 = fmaf(v, att_a[hh], s1);
    s2 = fmaf(v, att_a[H + hh], s2);
  }
#pragma unroll
  for (int off = 16; off > 0; off >>= 1) {
    s1 += __shfl_down(s1, off, 32);
    s2 += __shfl_down(s2, off, 32);
  }
  if (lane == 0) { f1[row] = s1; f2[row] = s2; }
}

// ---------------------------------------------------------------------------
// attention[i,j] = softmax_j( leaky_relu(f1[i] + f2[j]) )   (one block per row)
// f2 staged once in LDS (<=4KB) so the three passes hit ds_load, not global.
// ---------------------------------------------------------------------------
__global__ void __launch_bounds__(256) softmax_rows(
    const float* __restrict__ f1, const float* __restrict__ f2,
    float* __restrict__ attn, int Nn) {
  const int i    = (int)blockIdx.x;
  const int tid  = (int)threadIdx.x;
  const int lane = tid & 31;
  const int wid  = tid >> 5;
  const int nw   = (int)(blockDim.x >> 5);
  __shared__ float sf2[1024];
  __shared__ float red[8];
  __shared__ float bc;

  const float* __restrict__ f2p = f2;
  if (Nn <= 1024) {
    for (int j = tid; j < Nn; j += (int)blockDim.x) sf2[j] = f2[j];
    f2p = sf2;
  }
  __syncthreads();

  const float fi = f1[i];

  float lmax = -3.402823466e38f;
  for (int j = tid; j < Nn; j += (int)blockDim.x) {
    float e = fi + f2p[j];
    e = (e > 0.f) ? e : NEG_SLOPE * e;
    lmax = fmaxf(lmax, e);
  }
#pragma unroll
  for (int off = 16; off > 0; off >>= 1) lmax = fmaxf(lmax, __shfl_xor(lmax, off, 32));
  if (lane == 0) red[wid] = lmax;
  __syncthreads();
  if (tid == 0) {
    float mx = red[0];
    for (int w = 1; w < nw; ++w) mx = fmaxf(mx, red[w]);
    bc = mx;
  }
  __syncthreads();
  const float rmax = bc;

  float lsum = 0.f;
  for (int j = tid; j < Nn; j += (int)blockDim.x) {
    float e = fi + f2p[j];
    e = (e > 0.f) ? e : NEG_SLOPE * e;
    lsum += __expf(e - rmax);
  }
#pragma unroll
  for (int off = 16; off > 0; off >>= 1) lsum += __shfl_xor(lsum, off, 32);
  __syncthreads();                         // protect red/bc reuse
  if (lane == 0) red[wid] = lsum;
  __syncthreads();
  if (tid == 0) {
    float s = 0.f;
    for (int w = 0; w < nw; ++w) s += red[w];
    bc = 1.0f / s;
  }
  __syncthreads();
  const float inv = bc;

  float* __restrict__ out = attn + (size_t)i * Nn;
  for (int j = tid; j < Nn; j += (int)blockDim.x) {
    float e = fi + f2p[j];
    e = (e > 0.f) ? e : NEG_SLOPE * e;
    out[j] = __expf(e - rmax) * inv;
  }
}

// ---------------------------------------------------------------------------
// out[b,n] = sum_h relu(img_proj[b,h] + sem_proj[n,h] + sem_b[h]) * fc_w[h] + fc_b
// One block per b; img row + bias and fc_w staged in LDS, sem rows from L2.
// (relu sits inside the contraction -> inherently VALU, not matmul)
// ---------------------------------------------------------------------------
__global__ void __launch_bounds__(256) final_kernel(
    const float* __restrict__ img_proj, const float* __restrict__ sem_proj,
    const float* __restrict__ sem_b, const float* __restrict__ fc_w,
    const float* __restrict__ fc_b, float* __restrict__ out,
    int Nn, int H) {
  __shared__ float simg[1024];
  __shared__ float sw[1024];
  const int b = (int)blockIdx.x;
  for (int hh = (int)threadIdx.x; hh < H; hh += (int)blockDim.x) {
    simg[hh] = img_proj[(size_t)b * H + hh] + sem_b[hh];
    sw[hh]   = fc_w[hh];
  }
  __syncthreads();
  const float bias = fc_b[0];
  for (int n = (int)threadIdx.x; n < Nn; n += (int)blockDim.x) {
    const float* __restrict__ srow = sem_proj + (size_t)n * H;
    float s = 0.f;
#pragma unroll 8
    for (int hh = 0; hh < H; ++hh) {
      float v = simg[hh] + srow[hh];
      s = fmaf(fmaxf(v, 0.f), sw[hh], s);
    }
    out[(size_t)b * Nn + n] = s + bias;
  }
}

// ---------------------------------------------------------------------------
extern "C" void kernel_launch(void* const* d_in, const int* in_sizes, int n_in,
                              void* d_out, int out_size, void* d_ws, size_t ws_size,
                              hipStream_t stream) {
  (void)n_in; (void)out_size; (void)ws_size;
  const float* image_feats = (const float*)d_in[0];
  const float* attributes  = (const float*)d_in[1];
  const float* att_w       = (const float*)d_in[2];
  const float* att_a       = (const float*)d_in[3];
  const float* img_w       = (const float*)d_in[4];
  const float* sem_w       = (const float*)d_in[5];
  const float* sem_b       = (const float*)d_in[6];
  const float* fc_w        = (const float*)d_in[7];
  const float* fc_b        = (const float*)d_in[8];

  // Derive dims from flat sizes (B=256, N=1000, A=I=H=512 for the reference).
  const int H  = in_sizes[3] / 2;          // att_a is (2H, 1)
  const int A  = in_sizes[2] / H;          // att_w is (A, H)
  const int Nn = in_sizes[1] / A;          // attributes is (N, A)
  const int I  = in_sizes[4] / H;          // img_w is (I, H)
  const int Bb = in_sizes[0] / I;          // image_feats is (B, I)

  float* ws = (float*)d_ws;
  size_t o = 0;
  float* att_h    = ws + o; o += (size_t)Nn * H;
  float* attn     = ws + o; o += (size_t)Nn * Nn;
  float* att_outs = ws + o; o += (size_t)Nn * H;
  float* img_proj = ws + o; o += (size_t)Bb * H;
  float* f1       = ws + o; o += (size_t)Nn;
  float* f2       = ws + o; o += (size_t)Nn;
  float* sem_proj = attn;                  // attention is dead after att_outs

  float* outp = (float*)d_out;

  auto gemm = [&](const float* Am, const float* Bm, float* Dm, int M, int N, int K) {
    const int tM = (M + 31) / 32;
    const int tN = N / 32;
    const int waves  = tM * tN;
    const int blocks = (waves + 7) / 8;    // 8 wave32 per 256-thread block
    hipLaunchKernelGGL(wmma_gemm_f32, dim3(blocks), dim3(256), 0, stream,
                       Am, Bm, Dm, M, N, K);
  };

  // 1) att_h = attributes @ att_w                    (N x H, K = A)
  gemm(attributes, att_w, att_h, Nn, H, A);
  // 2) f1 = att_h @ a1, f2 = att_h @ a2              (rank-1 structure of e)
  hipLaunchKernelGGL(f1f2_kernel, dim3((Nn + 7) / 8), dim3(256), 0, stream,
                     att_h, att_a, f1, f2, Nn, H);
  // 3) attention = softmax_rows(leaky_relu(f1[i] + f2[j]))
  hipLaunchKernelGGL(softmax_rows, dim3(Nn), dim3(256), 0, stream,
                     f1, f2, attn, Nn);
  // 4) att_outs = attention @ att_h                  (N x H, K = N)
  gemm(attn, att_h, att_outs, Nn, H, Nn);
  // 5) img_proj = image_feats @ img_w                (B x H, K = I)
  gemm(image_feats, img_w, img_proj, Bb, H, I);
  // 6) sem_proj = att_outs @ sem_w                   (N x H, K = A)
  gemm(att_outs, sem_w, sem_proj, Nn, H, A);
  // 7) fused relu-dot epilogue
  hipLaunchKernelGGL(final_kernel, dim3(Bb), dim3(256), 0, stream,
                     img_proj, sem_proj, sem_b, fc_w, fc_b, outp, Nn, H);
}